// TransMatch_EX_44100724195726
// MI455X (gfx1250) — compile-verified
//
#include <hip/hip_runtime.h>
#include <hip/hip_bf16.h>

#define AGG_PARAM 1.0f

typedef float v2f __attribute__((ext_vector_type(2)));
typedef float v8f __attribute__((ext_vector_type(8)));

static constexpr int Bsz = 2048;   // batch
static constexpr int Dm  = 128;    // embedding dim
static constexpr int Sn  = 64;     // neighbor samples
static constexpr int Kt  = 20;     // topk

// ---------------------------------------------------------------------------
// Kernel 1: neighbor gather + masked-mean aggregation.
// One block per output row r = b*2 + i (4096 blocks), 128 threads = one dim d.
// Writes X1[r, 0:128]   = ent_agg
//        X1[r, 128:256] = AGG_PARAM * edge_agg      (ready for W_le GEMM)
// ---------------------------------------------------------------------------
__global__ void __launch_bounds__(128)
tm_gather_agg(const float* __restrict__ entity_emb,
              const float* __restrict__ relation_emb,
              const int*  __restrict__ entity_pairs,
              const int*  __restrict__ train_edges,
              const int*  __restrict__ entity2edges,
              const int*  __restrict__ edge2relation,
              const int*  __restrict__ edge2entities,
              float* __restrict__ X1)
{
    const int r  = blockIdx.x;          // 0 .. 2B-1
    const int b  = r >> 1;
    const int d  = threadIdx.x;         // 0 .. 127
    const int ent = entity_pairs[r];
    const int te  = train_edges[b];
    const int* __restrict__ edges = entity2edges + (size_t)ent * Sn;

    float acc_edge = 0.f, acc_ent = 0.f;
    int cnt = 0;
    for (int s = 0; s < Sn; ++s) {
        const int e   = edges[s];
        const int rel = edge2relation[e];
        const int e0  = edge2entities[2 * e];
        const int e1  = edge2entities[2 * e + 1];
        const float ev = relation_emb[(size_t)rel * Dm + d];
        const float nv = 0.5f * (entity_emb[(size_t)e0 * Dm + d] +
                                 entity_emb[(size_t)e1 * Dm + d]);
        if (e != te) { acc_edge += ev; acc_ent += nv; ++cnt; }
    }
    const float nei = (float)(cnt ? cnt : 1);
    X1[(size_t)r * 256 + d]      = acc_ent / nei;
    X1[(size_t)r * 256 + Dm + d] = AGG_PARAM * acc_edge / nei;
}

// ---------------------------------------------------------------------------
// Kernel 2 (generic): C[M,128] = A[M,Ka] @ W[Ka,128] + bias[128]  (fp32 WMMA)
// 256 threads = 8 waves; each wave computes one 16x128 strip of C using
// V_WMMA_F32_16X16X4_F32 (K stepped by 4). M must be a multiple of 128.
// ---------------------------------------------------------------------------
__global__ void __launch_bounds__(256)
tm_gemm_n128(const float* __restrict__ A, const float* __restrict__ W,
             const float* __restrict__ bias, float* __restrict__ C,
             int M, int Ka)
{
    const int wave = threadIdx.x >> 5;
    const int lane = threadIdx.x & 31;
    const int lh   = lane & 15;         // position within 16
    const int hi   = lane >> 4;         // 0 or 1 (K-halves for A/B operands)
    const int m0   = (blockIdx.x * 8 + wave) * 16;
    if (m0 >= M) return;

    v8f acc[8] = {};                    // 8 col tiles x (16x16 f32) = 16x128
    const float* __restrict__ Arow = A + (size_t)(m0 + lh) * Ka;

    for (int k = 0; k < Ka; k += 4) {
        // A operand: lanes 0-15 -> (K=k, k+1), lanes 16-31 -> (K=k+2, k+3)
        v2f a;
        a.x = Arow[k + 2 * hi];
        a.y = Arow[k + 2 * hi + 1];
#pragma unroll
        for (int t = 0; t < 8; ++t) {
            const int col = t * 16 + lh;
            v2f bb;
            bb.x = W[(size_t)(k + 2 * hi) * 128 + col];
            bb.y = W[(size_t)(k + 2 * hi + 1) * 128 + col];
            acc[t] = __builtin_amdgcn_wmma_f32_16x16x4_f32(
                false, a, false, bb, (short)0, acc[t], false, false);
        }
    }
    // C/D layout: VGPR j -> M = m0 + j (lanes 0-15) / m0 + j + 8 (lanes 16-31)
#pragma unroll
    for (int t = 0; t < 8; ++t) {
        const int col = t * 16 + lh;
        const float bv = bias ? bias[col] : 0.f;
#pragma unroll
        for (int j = 0; j < 8; ++j) {
            const int row = m0 + j + hi * 8;
            C[(size_t)row * 128 + col] = acc[t][j] + bv;
        }
    }
}

// ---------------------------------------------------------------------------
// Kernel 3: pack X2[b] = [ relation_emb[relations[b]] | AGG * nv[b, 0:256] ]
// ---------------------------------------------------------------------------
__global__ void __launch_bounds__(128)
tm_pack_x2(const float* __restrict__ relation_emb,
           const int*  __restrict__ relations,
           const float* __restrict__ NV,          // [2B,128] rows r=2b+i
           float* __restrict__ X2)                // [B,384]
{
    const int b = blockIdx.x;
    const int d = threadIdx.x;
    X2[(size_t)b * 384 + d]       = relation_emb[(size_t)relations[b] * Dm + d];
    X2[(size_t)b * 384 + 128 + d] = AGG_PARAM * NV[(size_t)(2 * b)     * Dm + d];
    X2[(size_t)b * 384 + 256 + d] = AGG_PARAM * NV[(size_t)(2 * b + 1) * Dm + d];
}

// ---------------------------------------------------------------------------
// Kernel 4: fused k/v projection (WMMA, staged in LDS) + softmax attention +
// final scorer. One block per batch element b; 256 threads = 8 waves.
//   waves 0-3: k = topk[b] @ W_k + b_k  (rows padded 20->32, 2x16 row tiles,
//              2x64 col halves);  waves 4-7: v likewise.
// ---------------------------------------------------------------------------
__global__ void __launch_bounds__(256)
tm_attn_score(const float* __restrict__ topk_emb,   // [B,K,128]
              const float* __restrict__ W_k, const float* __restrict__ b_k,
              const float* __restrict__ W_v, const float* __restrict__ b_v,
              const float* __restrict__ Q,          // [B,128]
              const float* __restrict__ AGGO,       // [B,128]
              const float* __restrict__ W_s, const float* __restrict__ b_s,
              float* __restrict__ out)              // [B]
{
    __shared__ float kv[2][32][128];                // 32 KB: k then v tiles
    __shared__ float sc[Kt];
    __shared__ float wgt[Kt];
    __shared__ float red[128];

    const int bb   = blockIdx.x;
    const int tid  = threadIdx.x;
    const int wave = tid >> 5;
    const int lane = tid & 31;
    const int lh   = lane & 15;
    const int hi   = lane >> 4;

    const int mat   = wave >> 2;          // 0 = k, 1 = v
    const int mtile = (wave >> 1) & 1;    // row tile (rows 0-15 / 16-31)
    const int chalf = wave & 1;           // col half (0-63 / 64-127)
    const float* __restrict__ Wm = mat ? W_v : W_k;
    const float* __restrict__ bm = mat ? b_v : b_k;

    // clamp padded rows (20..31) into the buffer; values unused downstream
    int rowA = bb * Kt + mtile * 16 + lh;
    if (rowA > Bsz * Kt - 1) rowA = Bsz * Kt - 1;
    const float* __restrict__ Arow = topk_emb + (size_t)rowA * Dm;

    v8f acc[4] = {};
    for (int k = 0; k < Dm; k += 4) {
        v2f a;
        a.x = Arow[k + 2 * hi];
        a.y = Arow[k + 2 * hi + 1];
#pragma unroll
        for (int t = 0; t < 4; ++t) {
            const int col = chalf * 64 + t * 16 + lh;
            v2f bv2;
            bv2.x = Wm[(size_t)(k + 2 * hi) * 128 + col];
            bv2.y = Wm[(size_t)(k + 2 * hi + 1) * 128 + col];
            acc[t] = __builtin_amdgcn_wmma_f32_16x16x4_f32(
                false, a, false, bv2, (short)0, acc[t], false, false);
        }
    }
#pragma unroll
    for (int t = 0; t < 4; ++t) {
        const int col = chalf * 64 + t * 16 + lh;
        const float bv = bm[col];
#pragma unroll
        for (int j = 0; j < 8; ++j) {
            const int row = mtile * 16 + j + hi * 8;
            kv[mat][row][col] = acc[t][j] + bv;
        }
    }
    __syncthreads();

    // scores_t = (q . k_t) / sqrt(D)
    const float* __restrict__ qrow = Q + (size_t)bb * Dm;
    if (tid < Kt) {
        float s = 0.f;
        for (int d = 0; d < Dm; ++d) s += qrow[d] * kv[0][tid][d];
        sc[tid] = s * 0.088388347648318440550f;    // 1/sqrt(128)
    }
    __syncthreads();

    if (tid == 0) {
        float mx = sc[0];
        for (int t = 1; t < Kt; ++t) mx = fmaxf(mx, sc[t]);
        float ssum = 0.f;
        for (int t = 0; t < Kt; ++t) { wgt[t] = __expf(sc[t] - mx); ssum += wgt[t]; }
        const float inv = 1.f / ssum;
        for (int t = 0; t < Kt; ++t) wgt[t] *= inv;
    }
    __syncthreads();

    // attn + fuse + scorer partial products
    if (tid < Dm) {
        float a = 0.f;
        for (int t = 0; t < Kt; ++t) a += wgt[t] * kv[1][t][tid];
        const float fused = AGGO[(size_t)bb * Dm + tid] + a;
        red[tid] = fused * W_s[tid];
    }
    __syncthreads();
    for (int off = 64; off > 0; off >>= 1) {
        if (tid < off) red[tid] += red[tid + off];
        __syncthreads();
    }
    if (tid == 0) out[bb] = red[0] + b_s[0];
}

// ---------------------------------------------------------------------------
extern "C" void kernel_launch(void* const* d_in, const int* in_sizes, int n_in,
                              void* d_out, int out_size, void* d_ws, size_t ws_size,
                              hipStream_t stream)
{
    const float* entity_emb   = (const float*)d_in[0];
    const float* relation_emb = (const float*)d_in[1];
    const float* W_le = (const float*)d_in[2];
    const float* b_le = (const float*)d_in[3];
    const float* W_l  = (const float*)d_in[4];
    const float* b_l  = (const float*)d_in[5];
    const float* W_q  = (const float*)d_in[6];
    const float* b_q  = (const float*)d_in[7];
    const float* W_k  = (const float*)d_in[8];
    const float* b_k  = (const float*)d_in[9];
    const float* W_v  = (const float*)d_in[10];
    const float* b_v  = (const float*)d_in[11];
    const float* W_s  = (const float*)d_in[12];
    const float* b_s  = (const float*)d_in[13];
    const float* user_emb = (const float*)d_in[14];
    const float* topk_emb = (const float*)d_in[15];
    const int* entity_pairs  = (const int*)d_in[16];
    const int* relations     = (const int*)d_in[17];
    const int* train_edges   = (const int*)d_in[18];
    const int* entity2edges  = (const int*)d_in[19];
    const int* edge2relation = (const int*)d_in[20];
    const int* edge2entities = (const int*)d_in[21];
    float* out = (float*)d_out;

    float* ws   = (float*)d_ws;
    float* X1   = ws;                      // [4096,256]
    float* NV   = X1 + 4096 * 256;         // [4096,128]
    float* X2   = NV + 4096 * 128;         // [2048,384]
    float* AGGO = X2 + 2048 * 384;         // [2048,128]
    float* Qb   = AGGO + 2048 * 128;       // [2048,128]

    // 1) gather + aggregate -> X1
    tm_gather_agg<<<Bsz * 2, 128, 0, stream>>>(
        entity_emb, relation_emb, entity_pairs, train_edges,
        entity2edges, edge2relation, edge2entities, X1);

    // 2) nv = X1 @ W_le + b_le        [4096,256]x[256,128]
    tm_gemm_n128<<<4096 / 128, 256, 0, stream>>>(X1, W_le, b_le, NV, 4096, 256);

    // 3) X2 = [self_vec | AGG*nv]
    tm_pack_x2<<<Bsz, 128, 0, stream>>>(relation_emb, relations, NV, X2);

    // 4) agg_out = X2 @ W_l + b_l     [2048,384]x[384,128]
    tm_gemm_n128<<<2048 / 128, 256, 0, stream>>>(X2, W_l, b_l, AGGO, 2048, 384);

    // 5) q = user_emb @ W_q + b_q     [2048,128]x[128,128]
    tm_gemm_n128<<<2048 / 128, 256, 0, stream>>>(user_emb, W_q, b_q, Qb, 2048, 128);

    // 6) fused k/v projection + attention + scorer -> out[B]
    tm_attn_score<<<Bsz, 256, 0, stream>>>(
        topk_emb, W_k, b_k, W_v, b_v, Qb, AGGO, W_s, b_s, out);
}